// SAGE_9723805958531
// MI455X (gfx1250) — compile-verified
//
#include <hip/hip_runtime.h>
#include <math.h>

// ---------------------------------------------------------------------------
// GraphSAGE (gcn aggregator), 3 layers, on MI455X (gfx1250, wave32).
//
// Strategy:
//  * Build CSR (edges sorted by dst) once per launch -> per-layer aggregation
//    becomes an atomic-free gather that runs out of the 192MB L2 (features are
//    25.6MB, fully resident).
//  * GEMM uses V_WMMA_F32_16X16X4_F32 (full fp32 matrix pipe). GEMM is only
//    ~4 GFLOP total, so precision is free and wmma throughput is irrelevant.
// ---------------------------------------------------------------------------

typedef __attribute__((ext_vector_type(2))) float v2f;
typedef __attribute__((ext_vector_type(8))) float v8f;

#define DIN 128
#define LDS_STRIDE 132   // 128 + 4 pad floats -> conflict-free b64 LDS reads

// ---- CSR build ------------------------------------------------------------

__global__ __launch_bounds__(256) void deg_kernel(const int* __restrict__ dst,
                                                  int* __restrict__ counts, int E) {
  int e = blockIdx.x * 256 + threadIdx.x;
  if (e < E) atomicAdd(&counts[dst[e]], 1);
}

__global__ __launch_bounds__(1024) void scan_kernel(const int* __restrict__ counts,
                                                    int* __restrict__ rowptr, int n) {
  __shared__ int tmp[1024];
  __shared__ int carry_s;
  if (threadIdx.x == 0) { carry_s = 0; rowptr[0] = 0; }
  __syncthreads();
  for (int base = 0; base < n; base += 1024) {
    int i = base + (int)threadIdx.x;
    int v = (i < n) ? counts[i] : 0;
    tmp[threadIdx.x] = v;
    __syncthreads();
    for (int off = 1; off < 1024; off <<= 1) {
      int t = (threadIdx.x >= (unsigned)off) ? tmp[threadIdx.x - off] : 0;
      __syncthreads();
      tmp[threadIdx.x] += t;
      __syncthreads();
    }
    int incl = tmp[threadIdx.x] + carry_s;   // reads carry_s (old value)
    if (i < n) rowptr[i + 1] = incl;
    __syncthreads();                          // protect carry_s RAW
    if (threadIdx.x == 1023) carry_s += tmp[1023];
    __syncthreads();
  }
}

__global__ __launch_bounds__(256) void scatter_kernel(const int* __restrict__ src,
                                                      const int* __restrict__ dst,
                                                      const int* __restrict__ rowptr,
                                                      int* __restrict__ cursor,
                                                      int* __restrict__ esorted, int E) {
  int e = blockIdx.x * 256 + threadIdx.x;
  if (e < E) {
    int d = dst[e];
    int p = atomicAdd(&cursor[d], 1);
    esorted[rowptr[d] + p] = src[e];
  }
}

// ---- Per-layer neighbor gather + normalize --------------------------------
// One wave32 per node; lane l owns features [4l, 4l+3] as a float4.
// Per-edge the wave reads 512 consecutive bytes (L2-resident).

__global__ __launch_bounds__(256) void gather_kernel(const float* __restrict__ in,
                                                     const int* __restrict__ rowptr,
                                                     const int* __restrict__ esorted,
                                                     float* __restrict__ hn, int N) {
  int wid  = (int)((blockIdx.x * 256u + threadIdx.x) >> 5);
  int lane = threadIdx.x & 31;
  if (wid >= N) return;
  int beg = rowptr[wid];
  int end = rowptr[wid + 1];

  float4 acc = ((const float4*)(in + (size_t)wid * DIN))[lane];  // self term
  int s_cur = (beg < end) ? esorted[beg] : 0;
  for (int e = beg; e < end; ++e) {
    int s_nxt = (e + 1 < end) ? esorted[e + 1] : 0;   // pipeline the index load
    float4 v = ((const float4*)(in + (size_t)s_cur * DIN))[lane];
    acc.x += v.x; acc.y += v.y; acc.z += v.z; acc.w += v.w;
    s_cur = s_nxt;
  }
  float rinv = 1.0f / (float)(end - beg + 1);
  float4 o;
  o.x = acc.x * rinv; o.y = acc.y * rinv; o.z = acc.z * rinv; o.w = acc.w * rinv;
  ((float4*)(hn + (size_t)wid * DIN))[lane] = o;
}

// ---- Fused GEMM + bias + activation with V_WMMA_F32_16X16X4_F32 -----------
// out[node, n] = sum_k hn[node, k] * W[n, k] + b[n], then ReLU/sigmoid.
// Fragment layouts per CDNA5 ISA 7.12.2:
//   A (16x4 f32):  element (m,k) -> lane = m + 16*(k>>1), vgpr = k&1
//   B (4x16 f32):  element (k,n) -> lane = n + 16*(k>>1), vgpr = k&1
//   C/D (16x16):   vgpr r -> M = r + 8*(lane>>4), N = lane&15

template <int DOUT, int ACT>
__global__ __launch_bounds__(256) void gemm_kernel(const float* __restrict__ hn,
                                                   const float* __restrict__ W,
                                                   const float* __restrict__ bias,
                                                   float* __restrict__ out, int N) {
  constexpr int CT   = DOUT / 16;   // column tiles across waves
  constexpr int RT   = 8 / CT;      // row tiles per block (8 waves/block)
  constexpr int ROWS = RT * 16;
  __shared__ float lds[ROWS * LDS_STRIDE];

  const int row0 = blockIdx.x * ROWS;

  // Stage a ROWS x 128 tile of hn into padded LDS (zero-fill past N).
  for (int i = threadIdx.x; i < ROWS * 32; i += 256) {
    int r = i >> 5, c4 = i & 31;
    int node = row0 + r;
    float4 v;
    if (node < N) v = ((const float4*)(hn + (size_t)node * DIN))[c4];
    else { v.x = 0.f; v.y = 0.f; v.z = 0.f; v.w = 0.f; }
    *(float4*)&lds[r * LDS_STRIDE + c4 * 4] = v;
  }
  __syncthreads();

  const int w    = threadIdx.x >> 5;
  const int lane = threadIdx.x & 31;
  const int half = lane >> 4;
  const int l16  = lane & 15;
  const int m0   = (w / CT) * 16;
  const int n0   = (w % CT) * 16;

  v8f acc = {0.f, 0.f, 0.f, 0.f, 0.f, 0.f, 0.f, 0.f};

  const float* arow = &lds[(m0 + l16) * LDS_STRIDE + 2 * half];       // 8B aligned
  const float* brow = W + (size_t)(n0 + l16) * DIN + 2 * half;        // 8B aligned

#pragma unroll
  for (int kb = 0; kb < DIN; kb += 4) {
    v2f a, b;
    a.x = arow[kb]; a.y = arow[kb + 1];   // ds_load_b64
    b.x = brow[kb]; b.y = brow[kb + 1];   // global_load_b64 (W is L2/WGP$ hot)
    acc = __builtin_amdgcn_wmma_f32_16x16x4_f32(
        /*neg_a=*/false, a, /*neg_b=*/false, b,
        /*c_mod=*/(short)0, acc, /*reuse_a=*/false, /*reuse_b=*/false);
  }

  float bv = bias[n0 + l16];
#pragma unroll
  for (int r = 0; r < 8; ++r) {
    int m = m0 + r + 8 * half;
    int node = row0 + m;
    if (node < N) {
      float v = acc[r] + bv;
      if (ACT == 0) v = v > 0.f ? v : 0.f;                  // ReLU
      else          v = 1.0f / (1.0f + __expf(-v));          // sigmoid
      out[(size_t)node * DOUT + (n0 + l16)] = v;
    }
  }
}

// ---------------------------------------------------------------------------

extern "C" void kernel_launch(void* const* d_in, const int* in_sizes, int n_in,
                              void* d_out, int out_size, void* d_ws, size_t ws_size,
                              hipStream_t stream) {
  (void)n_in; (void)out_size; (void)ws_size;
  const float* x  = (const float*)d_in[0];
  const int*   src = (const int*)d_in[1];
  const int*   dst = (const int*)d_in[2];
  const float* W1 = (const float*)d_in[3];
  const float* b1 = (const float*)d_in[4];
  const float* W2 = (const float*)d_in[5];
  const float* b2 = (const float*)d_in[6];
  const float* W3 = (const float*)d_in[7];
  const float* b3 = (const float*)d_in[8];
  const int N = in_sizes[0] / DIN;
  const int E = in_sizes[1];
  float* out = (float*)d_out;

  // Workspace carve-out (aligned to 256B): ~55 MB total.
  char* p = (char*)d_ws;
  auto carve = [&](size_t bytes) -> char* {
    char* q = p; p += (bytes + 255) & ~(size_t)255; return q;
  };
  int*   counts  = (int*)carve((size_t)N * 4);
  int*   rowptr  = (int*)carve((size_t)(N + 1) * 4);
  int*   cursor  = (int*)carve((size_t)N * 4);
  int*   esorted = (int*)carve((size_t)E * 4);
  float* hn      = (float*)carve((size_t)N * DIN * 4);
  float* z       = (float*)carve((size_t)N * DIN * 4);

  hipMemsetAsync(counts, 0, (size_t)N * 4, stream);
  hipMemsetAsync(cursor, 0, (size_t)N * 4, stream);

  const int eb = (E + 255) / 256;
  deg_kernel<<<eb, 256, 0, stream>>>(dst, counts, E);
  scan_kernel<<<1, 1024, 0, stream>>>(counts, rowptr, N);
  scatter_kernel<<<eb, 256, 0, stream>>>(src, dst, rowptr, cursor, esorted, E);

  const int gb = (N * 32 + 255) / 256;   // one wave32 per node

  // Layer 1: x -> hn -> z (ReLU)
  gather_kernel<<<gb, 256, 0, stream>>>(x, rowptr, esorted, hn, N);
  gemm_kernel<128, 0><<<(N + 15) / 16, 256, 0, stream>>>(hn, W1, b1, z, N);

  // Layer 2: z -> hn -> z (ReLU)
  gather_kernel<<<gb, 256, 0, stream>>>(z, rowptr, esorted, hn, N);
  gemm_kernel<128, 0><<<(N + 15) / 16, 256, 0, stream>>>(hn, W2, b2, z, N);

  // Layer 3: z -> hn -> out (sigmoid, DOUT=64)
  gather_kernel<<<gb, 256, 0, stream>>>(z, rowptr, esorted, hn, N);
  gemm_kernel<64, 1><<<(N + 31) / 32, 256, 0, stream>>>(hn, W3, b3, out, N);
}